// LabelPropagation_5282809774193
// MI455X (gfx1250) — compile-verified
//
#include <hip/hip_runtime.h>
#include <cstdint>

// Label propagation on MI455X (gfx1250).
//  - FP32 throughout (memory/atomic bound; matrix units irrelevant at 0.25 FLOP/B).
//  - Working set (~65 MB) is L2-resident (192 MB global L2): scatter uses
//    GLOBAL_ATOMIC_ADD_F32 resolved in L2.
//  - wave32: lane == class (C = 32), one wave handles one edge row per step.
//  - CDNA5 async path: edge indices double-buffered into LDS with
//    global_load_async_to_lds_b32 + s_wait_asynccnt.
//  - Elementwise passes vectorized to B128 (float4).

#define LP_ALPHA 0.9f
#define LP_OMA   0.1f   // 1 - alpha
#define LP_C     32
#define LP_K     10

// ---- CDNA5 async helpers ---------------------------------------------------

// Generic pointer to __shared__ is a flat address in the LDS aperture; its low
// 32 bits are the workgroup-relative LDS byte offset, which is exactly what
// the async-to-LDS instruction consumes in its VDST VGPR.
__device__ __forceinline__ unsigned lp_lds_off(const void* p) {
    return (unsigned)(unsigned long long)(uintptr_t)p;
}

__device__ __forceinline__ void lp_async_dword_to_lds(unsigned lds_byte_addr,
                                                      const void* gptr) {
    unsigned long long ga = (unsigned long long)(uintptr_t)gptr;
    asm volatile("global_load_async_to_lds_b32 %0, %1, off"
                 :: "v"(lds_byte_addr), "v"(ga)
                 : "memory");
}

__device__ __forceinline__ void lp_wait_async0() {
    asm volatile("s_wait_asynccnt 0x0" ::: "memory");
}

// ---- Scatter kernel (first in file so the async ops show in the disasm) ----

// agg[dst] += y[src] * normj[src]; one wave per edge row; edge indices staged
// into LDS via CDNA5 async loads, double-buffered across grid-stride chunks.
__global__ void __launch_bounds__(256)
lp_scatter(const float* __restrict__ y,
           const float* __restrict__ normj,
           const int*   __restrict__ src,
           const int*   __restrict__ dst,
           float* __restrict__ agg,
           int E) {
    __shared__ unsigned stage[8][2][2][32];   // [wave][buf][src/dst][lane]

    const int lane = threadIdx.x & 31;
    const int wv   = threadIdx.x >> 5;

    long long wgid    = (long long)blockIdx.x * 8 + wv;
    long long wstride = (long long)gridDim.x * 8;
    long long nchunks = ((long long)E + 31) >> 5;

    auto issue = [&](int buf, long long chunk) {
        long long e = (chunk << 5) + lane;
        if (e >= E) e = (long long)E - 1;          // clamp, never OOB
        lp_async_dword_to_lds(lp_lds_off(&stage[wv][buf][0][lane]), src + e);
        lp_async_dword_to_lds(lp_lds_off(&stage[wv][buf][1][lane]), dst + e);
    };

    int buf = 0;
    long long chunk = wgid;
    if (chunk < nchunks) issue(buf, chunk);

    for (; chunk < nchunks; chunk += wstride) {
        lp_wait_async0();                           // staged indices resident
        unsigned ms = stage[wv][buf][0][lane];
        unsigned md = stage[wv][buf][1][lane];

        long long nxt = chunk + wstride;            // prefetch next chunk while
        if (nxt < nchunks) issue(buf ^ 1, nxt);     // we do the atomic scatter

        long long base = chunk << 5;
        long long rem  = (long long)E - base;
        if (rem >= 32) {                            // common case: full chunk
#pragma unroll 8
            for (int k = 0; k < 32; ++k) {
                int s = __shfl((int)ms, k, 32);
                int d = __shfl((int)md, k, 32);
                float v = y[(long long)s * LP_C + lane] * normj[s];
                atomicAdd(&agg[(long long)d * LP_C + lane], v);
            }
        } else {
            for (int k = 0; k < (int)rem; ++k) {
                int s = __shfl((int)ms, k, 32);
                int d = __shfl((int)md, k, 32);
                float v = y[(long long)s * LP_C + lane] * normj[s];
                atomicAdd(&agg[(long long)d * LP_C + lane], v);
            }
        }
        buf ^= 1;
    }
}

// ---- Elementwise kernels (float4 / B128) -----------------------------------

// Per-quad (4 floats): y0 = mask ? labels : 0 ; init = (1-a)*y0 ; agg = 0.
// Also zeroes the per-node degree accumulators.
__global__ void lp_init4(const float4* __restrict__ labels,
                         const int*    __restrict__ mask,
                         float4* __restrict__ y0,
                         float4* __restrict__ initb,
                         float4* __restrict__ agg,
                         float*  __restrict__ normi,
                         float*  __restrict__ normj,
                         int N) {
    long long q  = (long long)blockIdx.x * blockDim.x + threadIdx.x;
    long long NQ = (long long)N * (LP_C / 4);       // quads
    if (q < NQ) {
        int row = (int)(q >> 3);                    // 8 quads per 32-wide row
        float4 l = labels[q];
        if (mask[row] == 0) l = make_float4(0.f, 0.f, 0.f, 0.f);
        y0[q] = l;
        initb[q] = make_float4(LP_OMA * l.x, LP_OMA * l.y,
                               LP_OMA * l.z, LP_OMA * l.w);
        agg[q] = make_float4(0.f, 0.f, 0.f, 0.f);
    }
    if (q < N) { normi[q] = 0.0f; normj[q] = 0.0f; }
}

// degree accumulation with float atomics (exact: degrees << 2^24)
__global__ void lp_deg(const int* __restrict__ src,
                       const int* __restrict__ dst,
                       float* __restrict__ normi,
                       float* __restrict__ normj,
                       int E) {
    long long e = (long long)blockIdx.x * blockDim.x + threadIdx.x;
    if (e < E) {
        atomicAdd(&normj[src[e]], 1.0f);   // out-degree
        atomicAdd(&normi[dst[e]], 1.0f);   // in-degree
    }
}

// norm = rsqrt(max(deg, 1))
__global__ void lp_norm(float* __restrict__ normi,
                        float* __restrict__ normj,
                        int N) {
    long long i = (long long)blockIdx.x * blockDim.x + threadIdx.x;
    if (i < N) {
        normi[i] = rsqrtf(fmaxf(normi[i], 1.0f));
        normj[i] = rsqrtf(fmaxf(normj[i], 1.0f));
    }
}

// Per-quad: y_out = clip(init + alpha*normi[row]*agg, 0, 1); re-zero agg.
__global__ void lp_update4(const float4* __restrict__ initb,
                           float4* __restrict__ agg,
                           const float* __restrict__ normi,
                           float4* __restrict__ yout,
                           int N) {
    long long q  = (long long)blockIdx.x * blockDim.x + threadIdx.x;
    long long NQ = (long long)N * (LP_C / 4);
    if (q >= NQ) return;
    int row = (int)(q >> 3);
    float s = LP_ALPHA * normi[row];
    float4 a  = agg[q];
    agg[q] = make_float4(0.f, 0.f, 0.f, 0.f);       // ready for next round
    float4 ib = initb[q];
    float4 v;
    v.x = fminf(fmaxf(fmaf(s, a.x, ib.x), 0.f), 1.f);
    v.y = fminf(fmaxf(fmaf(s, a.y, ib.y), 0.f), 1.f);
    v.z = fminf(fmaxf(fmaf(s, a.z, ib.z), 0.f), 1.f);
    v.w = fminf(fmaxf(fmaf(s, a.w, ib.w), 0.f), 1.f);
    yout[q] = v;
}

// ---- Host launch -----------------------------------------------------------

extern "C" void kernel_launch(void* const* d_in, const int* in_sizes, int n_in,
                              void* d_out, int out_size, void* d_ws, size_t ws_size,
                              hipStream_t stream) {
    const float* labels = (const float*)d_in[0];
    const int*   mask   = (const int*)d_in[1];
    const int*   src    = (const int*)d_in[2];
    const int*   dst    = (const int*)d_in[3];

    const int N = in_sizes[1];
    const int E = in_sizes[2];
    const long long NC = (long long)N * LP_C;
    const long long NQ = NC / 4;                    // float4 quads

    float* out   = (float*)d_out;
    float* ws    = (float*)d_ws;
    float* normi = ws;            // N
    float* normj = normi + N;     // N
    float* initb = normj + N;     // N*C   (16B-aligned: 2N*4 = 800000 B)
    float* agg   = initb + NC;    // N*C
    float* yodd  = agg + NC;      // N*C   (odd-step ping buffer)

    const int thr = 256;
    const int blocksNQ = (int)((NQ + thr - 1) / thr);
    const int blocksE  = (E + thr - 1) / thr;
    const int blocksN  = (N + thr - 1) / thr;

    lp_init4<<<blocksNQ, thr, 0, stream>>>((const float4*)labels, mask,
                                           (float4*)out, (float4*)initb,
                                           (float4*)agg, normi, normj, N);
    lp_deg<<<blocksE, thr, 0, stream>>>(src, dst, normi, normj, E);
    lp_norm<<<blocksN, thr, 0, stream>>>(normi, normj, N);

    const float* yprev = out;                      // y0 lives in d_out
    for (int k = 1; k <= LP_K; ++k) {
        float* ynext = (k & 1) ? yodd : out;       // K=10 even -> ends in d_out
        lp_scatter<<<2048, 256, 0, stream>>>(yprev, normj, src, dst, agg, E);
        lp_update4<<<blocksNQ, thr, 0, stream>>>((const float4*)initb,
                                                 (float4*)agg, normi,
                                                 (float4*)ynext, N);
        yprev = ynext;
    }
}